// SimpleGCN_86981677679168
// MI455X (gfx1250) — compile-verified
//
#include <hip/hip_runtime.h>
#include <hip/hip_bf16.h>

typedef __attribute__((ext_vector_type(2))) float v2f;
typedef __attribute__((ext_vector_type(8))) float v8f;

// ---------------------------------------------------------------- zero scratch
__global__ void gcn_zero_kernel(float* __restrict__ p, long long n) {
    long long i = (long long)blockIdx.x * blockDim.x + threadIdx.x;
    long long stride = (long long)gridDim.x * blockDim.x;
    for (; i < n; i += stride) p[i] = 0.0f;
}

// ---------------------------------------------------------------- degree pass
// deg_out[src] += 1 ; deg_in[dst] += 1   (edge_index is int64 [2,E])
__global__ void gcn_degree_kernel(const long long* __restrict__ ei, long long E,
                                  float* __restrict__ deg_out,
                                  float* __restrict__ deg_in) {
    long long e = (long long)blockIdx.x * blockDim.x + threadIdx.x;
    long long stride = (long long)gridDim.x * blockDim.x;
    for (; e < E; e += stride) {
        int s = (int)ei[e];
        int d = (int)ei[E + e];
        atomicAdd(&deg_out[s], 1.0f);
        atomicAdd(&deg_in[d], 1.0f);
    }
}

// ------------------------------------------------- norms (in place) + layer-1 msg
// norm = deg>0 ? rsqrt(max(deg,1)) : 0 ; m1 = x * norm_out
__global__ void gcn_norm_msg_kernel(const float* __restrict__ x,
                                    float* __restrict__ deg_out_to_norm,
                                    float* __restrict__ deg_in_to_norm,
                                    float* __restrict__ m1, int n_nodes) {
    int i = blockIdx.x * blockDim.x + threadIdx.x;
    if (i >= n_nodes) return;
    float dout = deg_out_to_norm[i];
    float nout = (dout > 0.0f) ? (1.0f / sqrtf(dout)) : 0.0f;
    float din  = deg_in_to_norm[i];
    float nin  = (din > 0.0f) ? (1.0f / sqrtf(din)) : 0.0f;
    deg_out_to_norm[i] = nout;
    deg_in_to_norm[i]  = nin;
    m1[i] = x[i] * nout;
}

// ---------------------------------------------------------------- edge scatter
// acc[dst] += msg[src]   (msg & acc arrays are L2-resident: 800KB each)
__global__ void gcn_scatter_kernel(const long long* __restrict__ ei, long long E,
                                   const float* __restrict__ msg,
                                   float* __restrict__ acc) {
    long long e = (long long)blockIdx.x * blockDim.x + threadIdx.x;
    long long stride = (long long)gridDim.x * blockDim.x;
    for (; e < E; e += stride) {
        int s = (int)ei[e];
        int d = (int)ei[E + e];
        atomicAdd(&acc[d], msg[s]);
    }
}

// -------------------------------------------------------- node MLP via WMMA f32
// For a tile of 16 nodes:  u[m] = s_agg[m]*norm_in[m]
//   H = u (16x1) x W1 (1x32)  via two V_WMMA_F32_16X16X4_F32 (rank-1: only K=0
//   column of A / K=0 row of B nonzero -> layout-robust, exact f32)
//   t[m] = norm_out[m] * sum_n relu(H[m,n]+b1[n]) * W2[n]
__global__ void gcn_mlp_wmma_kernel(const float* __restrict__ s_agg,
                                    const float* __restrict__ norm_in,
                                    const float* __restrict__ norm_out,
                                    const float* __restrict__ W1,
                                    const float* __restrict__ b1,
                                    const float* __restrict__ W2,
                                    float* __restrict__ t, int n_nodes) {
    const int lane = threadIdx.x & 31;
    const int half = lane >> 4;          // 0: lanes 0-15, 1: lanes 16-31
    const int l16  = lane & 15;
    const int wave   = (blockIdx.x * blockDim.x + threadIdx.x) >> 5;
    const int nwaves = (gridDim.x * blockDim.x) >> 5;

    // B operands: only row K=0 nonzero (lanes 0-15, first VGPR component)
    v2f bA = {0.0f, 0.0f};               // columns N = 0..15  of W1
    v2f bB = {0.0f, 0.0f};               // columns N = 16..31 of W1
    if (half == 0) { bA.x = W1[l16]; bB.x = W1[16 + l16]; }
    const float b1a = b1[l16],      b1b = b1[16 + l16];
    const float w2a = W2[l16],      w2b = W2[16 + l16];

    const int ntiles = (n_nodes + 15) >> 4;
    for (int tile = wave; tile < ntiles; tile += nwaves) {
        const int base = tile << 4;
        // A operand: only column K=0 nonzero (row m in lane m, lanes 0-15)
        int m  = base + l16;
        int mc = (m < n_nodes) ? m : (n_nodes - 1);   // clamp, keep EXEC full
        v2f a = {0.0f, 0.0f};
        if (half == 0) a.x = s_agg[mc] * norm_in[mc];

        v8f c0 = {}; v8f c1 = {};
        c0 = __builtin_amdgcn_wmma_f32_16x16x4_f32(false, a, false, bA,
                                                   (short)0, c0, false, false);
        c1 = __builtin_amdgcn_wmma_f32_16x16x4_f32(false, a, false, bB,
                                                   (short)0, c1, false, false);

        // C/D layout: element v of this lane is (M = v + 8*half, N = l16)
        float p[8];
#pragma unroll
        for (int v = 0; v < 8; ++v) {
            float h0 = c0[v] + b1a; h0 = h0 > 0.0f ? h0 : 0.0f;
            float h1 = c1[v] + b1b; h1 = h1 > 0.0f ? h1 : 0.0f;
            p[v] = h0 * w2a + h1 * w2b;
        }
        // reduce over the 16 lanes of each half (sum over width N)
#pragma unroll
        for (int v = 0; v < 8; ++v) {
            float s = p[v];
            s += __shfl_xor(s, 1, 32);
            s += __shfl_xor(s, 2, 32);
            s += __shfl_xor(s, 4, 32);
            s += __shfl_xor(s, 8, 32);
            p[v] = s;
        }
        if (l16 == 0) {                  // lanes 0 and 16 write 8 rows each
#pragma unroll
            for (int v = 0; v < 8; ++v) {
                int row = base + v + 8 * half;
                if (row < n_nodes) t[row] = norm_out[row] * p[v];
            }
        }
    }
}

// ---------------------------------------------------------------- graph pooling
__global__ void gcn_pool_kernel(const float* __restrict__ agg2,
                                const float* __restrict__ norm_in,
                                const long long* __restrict__ gid,
                                const float* __restrict__ b2,
                                float* __restrict__ gsum,
                                float* __restrict__ gcnt, int n_nodes) {
    int i = blockIdx.x * blockDim.x + threadIdx.x;
    if (i >= n_nodes) return;
    float val = agg2[i] * norm_in[i] + b2[0];
    int g = (int)gid[i];
    atomicAdd(&gsum[g], val);
    atomicAdd(&gcnt[g], 1.0f);
}

__global__ void gcn_finalize_kernel(const float* __restrict__ gsum,
                                    const float* __restrict__ gcnt,
                                    float* __restrict__ out, int n_graphs) {
    int g = blockIdx.x * blockDim.x + threadIdx.x;
    if (g >= n_graphs) return;
    out[g] = gsum[g] / fmaxf(gcnt[g], 1.0f);
}

// =============================================================================
extern "C" void kernel_launch(void* const* d_in, const int* in_sizes, int n_in,
                              void* d_out, int out_size, void* d_ws, size_t ws_size,
                              hipStream_t stream) {
    const float*     x   = (const float*)d_in[0];
    const long long* ei  = (const long long*)d_in[1];
    const long long* gid = (const long long*)d_in[2];
    const float*     W1  = (const float*)d_in[3];
    const float*     b1  = (const float*)d_in[4];
    const float*     W2  = (const float*)d_in[5];
    const float*     b2  = (const float*)d_in[6];
    float* out = (float*)d_out;

    const int       N = in_sizes[0];            // 200,000 nodes
    const long long E = (long long)in_sizes[1] / 2;  // 6,400,000 edges
    const int       G = out_size;               // 1000 graphs

    float* base     = (float*)d_ws;
    float* norm_out = base;                 // [N] deg_out -> norm_out (in place)
    float* norm_in  = base + (size_t)N;     // [N] deg_in  -> norm_in  (in place)
    float* s_agg    = base + 2*(size_t)N;   // [N] layer-1 scalar aggregate
    float* msg      = base + 3*(size_t)N;   // [N] m1, later reused for t
    float* agg2     = base + 4*(size_t)N;   // [N] layer-2 aggregate
    float* gsum     = base + 5*(size_t)N;   // [G]
    float* gcnt     = gsum + G;             // [G]

    const long long zn = 5LL * N + 2LL * G; // zero everything (msg is overwritten anyway)
    const int TB = 256;

    // 1. zero accumulators (fresh every call — no state across replays)
    gcn_zero_kernel<<<(int)((zn + TB - 1) / TB), TB, 0, stream>>>(base, zn);

    // 2. degrees via float atomics (exact for counts << 2^24)
    int eblocks = (int)((E + TB - 1) / TB);
    gcn_degree_kernel<<<eblocks, TB, 0, stream>>>(ei, E, norm_out, norm_in);

    // 3. norms (in place) + layer-1 scalar message m1 = x * norm_out
    gcn_norm_msg_kernel<<<(N + TB - 1) / TB, TB, 0, stream>>>(x, norm_out, norm_in, msg, N);

    // 4. layer-1 scatter: s_agg[dst] += m1[src]
    gcn_scatter_kernel<<<eblocks, TB, 0, stream>>>(ei, E, msg, s_agg);

    // 5. per-node MLP (1 -> 32 -> 1) on the matrix pipe; writes t into msg
    {
        int tiles  = (N + 15) / 16;                 // one wave per tile
        int blocks = (tiles * 32 + TB - 1) / TB;
        gcn_mlp_wmma_kernel<<<blocks, TB, 0, stream>>>(s_agg, norm_in, norm_out,
                                                       W1, b1, W2, msg, N);
    }

    // 6. layer-2 scatter: agg2[dst] += t[src]
    gcn_scatter_kernel<<<eblocks, TB, 0, stream>>>(ei, E, msg, agg2);

    // 7. per-graph mean readout
    gcn_pool_kernel<<<(N + TB - 1) / TB, TB, 0, stream>>>(agg2, norm_in, gid, b2,
                                                          gsum, gcnt, N);
    gcn_finalize_kernel<<<(G + TB - 1) / TB, TB, 0, stream>>>(gsum, gcnt, out, G);
}